// SoftCluster_adaptiveK_75462575391236
// MI455X (gfx1250) — compile-verified
//
#include <hip/hip_runtime.h>
#include <math.h>

// ---------------- problem constants ----------------
#define N_PTS 80000
#define DIM   256
#define KMAX  6
#define KPAD  16
#define TILE  128                       // points per tile (625 tiles exactly)
#define NTILES (N_PTS / TILE)           // 625
#define ABLK  160                       // persistent blocks for assign kernel
#define PPB   256                       // points per block for kmeans++ passes
#define BLK_PP ((N_PTS + PPB - 1) / PPB) // 313
#define TOL_F 1e-5f

typedef __attribute__((ext_vector_type(2))) float v2f;
typedef __attribute__((ext_vector_type(8))) float v8f;
typedef __attribute__((ext_vector_type(4))) int   v4i;

// ---------------- gfx1250 async global->LDS support (guarded) ----------------
#if defined(__has_builtin)
#if __has_builtin(__builtin_amdgcn_global_load_async_to_lds_b128)
#define HAVE_ASYNC_LDS 1
typedef __attribute__((address_space(1))) v4i g_v4i;   // global (AS1)
typedef __attribute__((address_space(3))) v4i l_v4i;   // LDS (AS3)
#endif
#endif

__device__ __forceinline__ void wait_async0()
{
#ifdef HAVE_ASYNC_LDS
#if __has_builtin(__builtin_amdgcn_s_wait_asynccnt)
    __builtin_amdgcn_s_wait_asynccnt(0);
#else
    asm volatile("s_wait_asynccnt 0" ::: "memory");
#endif
#endif
}

// Prefetch one 128x256 f32 tile (128KB) into an LDS buffer.
// Async path: 32 x global_load_async_to_lds_b128 per wave (ASYNCcnt <= 32 < 63).
__device__ __forceinline__ void prefetch_tile(const float4* __restrict__ xg4,
                                              float4* xsbuf, int tile, int tid)
{
    const float4* src = xg4 + (size_t)tile * (TILE * DIM / 4);
#ifdef HAVE_ASYNC_LDS
    #pragma unroll
    for (int i = 0; i < (TILE * DIM / 4) / 256; ++i) {
        __builtin_amdgcn_global_load_async_to_lds_b128(
            (g_v4i*)(src + i * 256 + tid),
            (l_v4i*)(xsbuf + i * 256 + tid),
            0, 0);
    }
#else
    #pragma unroll 4
    for (int i = 0; i < (TILE * DIM / 4) / 256; ++i)
        xsbuf[i * 256 + tid] = src[i * 256 + tid];
#endif
}

// ---------------- workspace layout (float offsets) ----------------
#define OFF_W      0                         // 16: gumbel-softmax weights (padded)
#define OFF_CSQ    16                        // 16: ||c_n||^2 (padded)
#define OFF_CTRL   32                        // 16: [0]=done(int) [1]=sel(int) [2]=target(float)
#define OFF_CENT   64                        // KPAD*DIM centers (rows >= k zeroed)
#define OFF_PREV   (OFF_CENT + KPAD*DIM)     // KPAD*DIM previous new_c
#define OFF_XSQ    (OFF_PREV + KPAD*DIM)     // N_PTS
#define OFF_DMIN   (OFF_XSQ + N_PTS)         // N_PTS
#define OFF_BSUM   (OFF_DMIN + N_PTS)        // 512 block sums (kmeans++)
#define OFF_OFFS   (OFF_BSUM + 512)          // 512 exclusive block offsets
#define OFF_SHIFT  (OFF_OFFS + 512)          // 16 per-cluster shift^2
#define OFF_PC     (OFF_SHIFT + 16)          // ABLK*KMAX*DIM partial center sums
#define OFF_PR     (OFF_PC + ABLK*KMAX*DIM)  // ABLK*KMAX partial r sums
// total ~416k floats (~1.67 MB)

// =======================================================================
// prep: xsq, gumbel-softmax weights, center 0 via u0, zero pads/prev
// =======================================================================
__global__ void __launch_bounds__(256) k_prep(const float* __restrict__ x,
                                              const float* __restrict__ logits,
                                              const float* __restrict__ gumbel,
                                              const float* __restrict__ uni,
                                              float* __restrict__ ws)
{
    const int tid  = threadIdx.x;
    const int lane = tid & 31;
    const int gw   = blockIdx.x * 8 + (tid >> 5);
    const int nw   = gridDim.x * 8;
    for (int p = gw; p < N_PTS; p += nw) {
        const float* xp = x + (size_t)p * DIM;
        float s = 0.f;
        #pragma unroll
        for (int j = 0; j < DIM / 32; ++j) { float v = xp[lane + 32 * j]; s += v * v; }
        #pragma unroll
        for (int m = 16; m >= 1; m >>= 1) s += __shfl_xor(s, m, 32);
        if (lane == 0) ws[OFF_XSQ + p] = s;
    }
    if (blockIdx.x == 0) {
        __shared__ float red[256];
        if (tid == 0) {
            float lg[KMAX], mx = -1e30f;
            for (int j = 0; j < KMAX; ++j) { lg[j] = logits[j] + gumbel[j]; mx = fmaxf(mx, lg[j]); }
            float sum = 0.f;
            for (int j = 0; j < KMAX; ++j) { lg[j] = __expf(lg[j] - mx); sum += lg[j]; }
            for (int j = 0; j < KMAX; ++j) ws[OFF_W + j] = lg[j] / sum;
            for (int j = KMAX; j < KPAD; ++j) ws[OFF_W + j] = 0.f;
            ((int*)(ws + OFF_CTRL))[0] = 0;  // done = false
        }
        int idx0 = (int)(uni[0] * (float)N_PTS);
        if (idx0 > N_PTS - 1) idx0 = N_PTS - 1;
        if (idx0 < 0) idx0 = 0;
        float c = x[(size_t)idx0 * DIM + tid];
        ws[OFF_CENT + tid] = c;
        for (int r = 1; r < KPAD; ++r) ws[OFF_CENT + r * DIM + tid] = 0.f;
        for (int r = 0; r < KPAD; ++r) ws[OFF_PREV + r * DIM + tid] = 0.f;
        red[tid] = c * c;
        __syncthreads();
        for (int s = 128; s >= 1; s >>= 1) { if (tid < s) red[tid] += red[tid + s]; __syncthreads(); }
        if (tid == 0) ws[OFF_CSQ + 0] = red[0];
        if (tid >= 1 && tid < KPAD) ws[OFF_CSQ + tid] = 0.f;
    }
}

// =======================================================================
// kmeans++ pass 1: dmin per point (nc centers), ordered block sums
// =======================================================================
__global__ void __launch_bounds__(256) k_dmin(const float* __restrict__ x,
                                              float* __restrict__ ws, int nc)
{
    __shared__ float cs[KMAX * DIM];
    __shared__ float wsum[8];
    const int tid = threadIdx.x;
    for (int i = tid; i < nc * DIM; i += 256) cs[i] = ws[OFF_CENT + i];
    __syncthreads();

    const int lane = tid & 31, w = tid >> 5;
    const int base = blockIdx.x * PPB + w * 32;
    float acc = 0.f;
    for (int q = 0; q < 32; ++q) {
        const int p = base + q;
        if (p >= N_PTS) break;
        const float* xp = x + (size_t)p * DIM;
        float dot[KMAX];
        for (int j = 0; j < KMAX; ++j) dot[j] = 0.f;
        for (int s8 = 0; s8 < DIM / 32; ++s8) {
            float xv = xp[lane + 32 * s8];
            for (int j = 0; j < nc; ++j) dot[j] += xv * cs[j * DIM + lane + 32 * s8];
        }
        for (int m = 16; m >= 1; m >>= 1)
            for (int j = 0; j < nc; ++j) dot[j] += __shfl_xor(dot[j], m, 32);
        if (lane == 0) {
            const float xq = ws[OFF_XSQ + p];
            float dmin = 1e30f;
            for (int j = 0; j < nc; ++j) {
                float d2 = xq + ws[OFF_CSQ + j] - 2.f * dot[j];
                dmin = fminf(dmin, sqrtf(fmaxf(d2, 1e-12f)));
            }
            ws[OFF_DMIN + p] = dmin;
            acc += dmin;
        }
    }
    if (lane == 0) wsum[w] = acc;
    __syncthreads();
    if (tid == 0) {
        float s = 0.f;
        for (int i = 0; i < 8; ++i) s += wsum[i];
        ws[OFF_BSUM + blockIdx.x] = s;
    }
}

// =======================================================================
// kmeans++ pass 2: scan block sums, compute target = u * total
// =======================================================================
__global__ void __launch_bounds__(512) k_scan(const float* __restrict__ uni,
                                              float* __restrict__ ws, int step)
{
    __shared__ float s0[512], s1[512];
    const int tid = threadIdx.x;
    const float v = (tid < BLK_PP) ? ws[OFF_BSUM + tid] : 0.f;
    s0[tid] = v;
    __syncthreads();
    float* src = s0; float* dst = s1;
    for (int off = 1; off < 512; off <<= 1) {
        float t = src[tid];
        if (tid >= off) t += src[tid - off];
        dst[tid] = t;
        __syncthreads();
        float* tmp = src; src = dst; dst = tmp;
    }
    if (tid < BLK_PP) ws[OFF_OFFS + tid] = src[tid] - v;   // exclusive offsets
    if (tid == 0) {
        const float total = src[BLK_PP - 1];
        ws[OFF_CTRL + 2] = uni[step] * total;              // target
        ((int*)(ws + OFF_CTRL))[1] = N_PTS - 1;            // sel idx (atomicMin)
    }
}

// =======================================================================
// kmeans++ pass 3: searchsorted via in-block scan + atomicMin
// =======================================================================
__global__ void __launch_bounds__(256) k_select(float* __restrict__ ws)
{
    __shared__ float s0[256], s1[256];
    const int tid = threadIdx.x;
    const int p = blockIdx.x * PPB + tid;
    const float v = (p < N_PTS) ? ws[OFF_DMIN + p] : 0.f;
    s0[tid] = v;
    __syncthreads();
    float* src = s0; float* dst = s1;
    for (int off = 1; off < 256; off <<= 1) {
        float t = src[tid];
        if (tid >= off) t += src[tid - off];
        dst[tid] = t;
        __syncthreads();
        float* tmp = src; src = dst; dst = tmp;
    }
    const float cdf = ws[OFF_OFFS + blockIdx.x] + src[tid];
    const float target = ws[OFF_CTRL + 2];
    if (p < N_PTS && cdf >= target && (cdf - v) < target)
        atomicMin(((int*)(ws + OFF_CTRL)) + 1, p);
}

// =======================================================================
// kmeans++ pass 4: commit chosen center
// =======================================================================
__global__ void __launch_bounds__(256) k_newcenter(const float* __restrict__ x,
                                                   float* __restrict__ ws, int row)
{
    __shared__ float red[256];
    const int tid = threadIdx.x;
    int sel = ((int*)(ws + OFF_CTRL))[1];
    if (sel < 0) sel = 0;
    if (sel > N_PTS - 1) sel = N_PTS - 1;
    const float c = x[(size_t)sel * DIM + tid];
    ws[OFF_CENT + row * DIM + tid] = c;
    red[tid] = c * c;
    __syncthreads();
    for (int s = 128; s >= 1; s >>= 1) { if (tid < s) red[tid] += red[tid + s]; __syncthreads(); }
    if (tid == 0) ws[OFF_CSQ + row] = red[0];
}

// =======================================================================
// main iteration: async double-buffered x tiles, WMMA dots -> softmax r
// -> partial r^T x. LDS: 2 x (128x256 f32) + centers + dot tiles + r
// =======================================================================
template <int K>
__global__ void __launch_bounds__(256) k_assign(const float* __restrict__ x,
                                                float* __restrict__ ws,
                                                float* __restrict__ rout)
{
    extern __shared__ float smem[];
    float* xsA  = smem;                        // TILE*DIM
    float* xsB  = xsA + TILE * DIM;            // TILE*DIM
    float* cs   = xsB + TILE * DIM;            // KPAD*DIM
    float* dt   = cs + KPAD * DIM;             // 8*16*17 (padded dot tiles)
    float* rs   = dt + 8 * 16 * 17;            // TILE*KMAX
    float* wv_s = rs + TILE * KMAX;            // 16
    float* cq_s = wv_s + 16;                   // 16

    if (((const int*)(ws + OFF_CTRL))[0] != 0) return;   // frozen (shift < TOL)

    const int tid = threadIdx.x;
    for (int i = tid; i < KPAD * DIM; i += 256) cs[i] = ws[OFF_CENT + i];
    if (tid < KPAD) { wv_s[tid] = ws[OFF_W + tid]; cq_s[tid] = ws[OFF_CSQ + tid]; }

    const int lane = tid & 31;
    const int wv   = tid >> 5;
    const int row  = lane & 15;            // A: M row / B: N col / C: N col
    const int half = lane >> 4;
    const int koff = half * 2;             // A/B K sub-offset per lane half

    float accC[K];
    #pragma unroll
    for (int n = 0; n < K; ++n) accC[n] = 0.f;
    float accR = 0.f;

    const float4* xg4 = (const float4*)x;

    // prefetch first tile into buffer A
    int cur = 0;
    prefetch_tile(xg4, (float4*)xsA, blockIdx.x, tid);

    for (int t = blockIdx.x; t < NTILES; t += gridDim.x) {
        const int pb = t * TILE;
        float* xs = cur ? xsB : xsA;
        float* xn = cur ? xsA : xsB;

        wait_async0();          // this wave's async writes into xs complete
        __syncthreads();        // all waves' tile data visible; prev reads of xn done

        const int tn = t + gridDim.x;
        if (tn < NTILES)        // overlap next tile DMA with compute
            prefetch_tile(xg4, (float4*)xn, tn, tid);

        // ---- per-wave 16x16 dot tile: 64 x v_wmma_f32_16x16x4_f32 ----
        v8f acc = {0.f, 0.f, 0.f, 0.f, 0.f, 0.f, 0.f, 0.f};
        const float* xrow = xs + (wv * 16 + row) * DIM + koff;  // A frag source
        const float* crow = cs + row * DIM + koff;              // B frag source (centers^T)
        #pragma unroll 8
        for (int kk = 0; kk < DIM; kk += 4) {
            v2f a = *(const v2f*)(xrow + kk);
            v2f b = *(const v2f*)(crow + kk);
            acc = __builtin_amdgcn_wmma_f32_16x16x4_f32(
                false, a, false, b, (short)0, acc, false, false);
        }
        {   // C layout: VGPR r, lane l -> M = r + 8*(l>>4), N = l&15
            float* dw = dt + wv * (16 * 17);
            #pragma unroll
            for (int r = 0; r < 8; ++r)
                dw[(r + 8 * half) * 17 + row] = acc[r];
        }
        __syncthreads();

        // ---- softmax responsibilities, one thread per point ----
        if (tid < TILE) {
            const int p = tid;
            const float xq = ws[OFF_XSQ + pb + p];
            const float* dw = dt + (p >> 4) * (16 * 17) + (p & 15) * 17;
            float dv[K];
            float dmin = 1e30f;
            #pragma unroll
            for (int n = 0; n < K; ++n) {
                float d2 = xq + cq_s[n] - 2.f * dw[n];
                float d = sqrtf(fmaxf(d2, 1e-12f));
                dv[n] = d;
                dmin = fminf(dmin, d);
            }
            float sum = 0.f;
            #pragma unroll
            for (int n = 0; n < K; ++n) { float e = wv_s[n] * __expf(dmin - dv[n]); dv[n] = e; sum += e; }
            const float inv = 1.f / sum;
            #pragma unroll
            for (int n = 0; n < K; ++n) {
                float r = dv[n] * inv;
                rs[p * KMAX + n] = r;
                rout[(size_t)(pb + p) * K + n] = r;
            }
        }
        __syncthreads();

        // ---- accumulate r^T x: thread = dim, loop points (LDS broadcast) ----
        for (int p = 0; p < TILE; ++p) {
            const float xv = xs[p * DIM + tid];
            #pragma unroll
            for (int n = 0; n < K; ++n) accC[n] += rs[p * KMAX + n] * xv;
        }
        if (tid < K)
            for (int p = 0; p < TILE; ++p) accR += rs[p * KMAX + tid];

        cur ^= 1;
        // no trailing barrier: next iteration's wait+barrier separates the
        // phase-4 reads of xs from the async writes that will refill it
    }
    __syncthreads();
    #pragma unroll
    for (int n = 0; n < K; ++n)
        ws[OFF_PC + ((size_t)blockIdx.x * KMAX + n) * DIM + tid] = accC[n];
    if (tid < K) ws[OFF_PR + blockIdx.x * KMAX + tid] = accR;
}

// =======================================================================
// reduce partials -> new centers, csq, shift^2  (one block per cluster)
// =======================================================================
__global__ void __launch_bounds__(256) k_reduce(float* __restrict__ ws,
                                                float* __restrict__ centers_out, int k)
{
    __shared__ float red[256];
    if (((const int*)(ws + OFF_CTRL))[0] != 0) return;
    const int n = blockIdx.x;
    const int tid = threadIdx.x;

    float pr = 0.f;
    for (int b = tid; b < ABLK; b += 256) pr += ws[OFF_PR + b * KMAX + n];
    red[tid] = pr;
    __syncthreads();
    for (int s = 128; s >= 1; s >>= 1) { if (tid < s) red[tid] += red[tid + s]; __syncthreads(); }
    const float sumr = red[0];
    __syncthreads();

    float sc = 0.f;
    for (int b = 0; b < ABLK; ++b) sc += ws[OFF_PC + ((size_t)b * KMAX + n) * DIM + tid];
    const float newc = sc / sumr;

    const float diff = newc - ws[OFF_PREV + n * DIM + tid];
    red[tid] = diff * diff;
    __syncthreads();
    for (int s = 128; s >= 1; s >>= 1) { if (tid < s) red[tid] += red[tid + s]; __syncthreads(); }
    if (tid == 0) ws[OFF_SHIFT + n] = red[0];
    __syncthreads();

    red[tid] = newc * newc;
    __syncthreads();
    for (int s = 128; s >= 1; s >>= 1) { if (tid < s) red[tid] += red[tid + s]; __syncthreads(); }
    if (tid == 0) ws[OFF_CSQ + n] = red[0];

    ws[OFF_PREV + n * DIM + tid] = newc;
    ws[OFF_CENT + n * DIM + tid] = newc;
    centers_out[n * DIM + tid]   = newc;
}

// =======================================================================
// finalize: done |= (||shift|| < TOL)
// =======================================================================
__global__ void k_finalize(float* __restrict__ ws, int k)
{
    if (threadIdx.x != 0) return;
    int* ctrl = (int*)(ws + OFF_CTRL);
    if (ctrl[0] != 0) return;
    float s = 0.f;
    for (int n = 0; n < k; ++n) s += ws[OFF_SHIFT + n];
    if (sqrtf(s) < TOL_F) ctrl[0] = 1;
}

// =======================================================================
extern "C" void kernel_launch(void* const* d_in, const int* in_sizes, int n_in,
                              void* d_out, int out_size, void* d_ws, size_t ws_size,
                              hipStream_t stream)
{
    const float* x      = (const float*)d_in[0];
    const float* logits = (const float*)d_in[1];
    const float* gumbel = (const float*)d_in[2];
    const float* uni    = (const float*)d_in[3];
    float* ws  = (float*)d_ws;
    float* out = (float*)d_out;

    int k = out_size / (N_PTS + DIM);   // out = centers[k,256] ++ r[80000,k]
    if (k < 1) k = 1;
    if (k > KMAX) k = KMAX;

    const size_t smem = (size_t)(2 * TILE * DIM + KPAD * DIM + 8 * 16 * 17 + TILE * KMAX + 32)
                        * sizeof(float);   // ~290 KB (<=320 KB/WGP LDS on CDNA5)

    const void* fptr = nullptr;
    switch (k) {
        case 1: fptr = (const void*)k_assign<1>; break;
        case 2: fptr = (const void*)k_assign<2>; break;
        case 3: fptr = (const void*)k_assign<3>; break;
        case 4: fptr = (const void*)k_assign<4>; break;
        case 5: fptr = (const void*)k_assign<5>; break;
        default: fptr = (const void*)k_assign<6>; break;
    }
    (void)hipFuncSetAttribute(fptr, hipFuncAttributeMaxDynamicSharedMemorySize, (int)smem);

    k_prep<<<256, 256, 0, stream>>>(x, logits, gumbel, uni, ws);

    for (int i = 1; i < k; ++i) {
        k_dmin<<<BLK_PP, 256, 0, stream>>>(x, ws, i);
        k_scan<<<1, 512, 0, stream>>>(uni, ws, i);
        k_select<<<BLK_PP, 256, 0, stream>>>(ws);
        k_newcenter<<<1, 256, 0, stream>>>(x, ws, i);
    }

    float* rout = out + (size_t)k * DIM;
    const int MAX_ITERS = 50;           // reference setup passes max_iters = 50
    for (int it = 0; it < MAX_ITERS; ++it) {
        switch (k) {
            case 1: k_assign<1><<<ABLK, 256, smem, stream>>>(x, ws, rout); break;
            case 2: k_assign<2><<<ABLK, 256, smem, stream>>>(x, ws, rout); break;
            case 3: k_assign<3><<<ABLK, 256, smem, stream>>>(x, ws, rout); break;
            case 4: k_assign<4><<<ABLK, 256, smem, stream>>>(x, ws, rout); break;
            case 5: k_assign<5><<<ABLK, 256, smem, stream>>>(x, ws, rout); break;
            default: k_assign<6><<<ABLK, 256, smem, stream>>>(x, ws, rout); break;
        }
        k_reduce<<<k, 256, 0, stream>>>(ws, out, k);
        k_finalize<<<1, 32, 0, stream>>>(ws, k);
    }
}